// CompressK_59691455480627
// MI455X (gfx1250) — compile-verified
//
#include <hip/hip_runtime.h>
#include <hip/hip_bf16.h>

// CompressK for MI455X (gfx1250).
// Memory-bound mean-pool (AI ~0.25 FLOP/byte -> pure bandwidth problem):
// stage each chunk's 32x2048B tile into LDS with a Tensor Data Mover DMA
// (tensor_load_to_lds, TENSORcnt), double-buffered across chunks so the DMA
// engine overlaps the LDS reduce; results leave via non-temporal b128 stores.

typedef __attribute__((ext_vector_type(4))) unsigned int v4u;
typedef __attribute__((ext_vector_type(8))) int          v8i;
typedef __attribute__((ext_vector_type(4))) int          v4i;
typedef __attribute__((ext_vector_type(4))) float        v4f;

#define ROW        512   // H*D = 4*128 floats per token row
#define ROW4       128   // float4 per token row
#define TILE_ROWS  32    // rows staged in LDS per TDM transfer (64 KB)
#define TILE_BYTES (TILE_ROWS * ROW * 4)
#define THREADS    128
#define MAXSEQ     256   // sequences whose metadata we cache in LDS

struct SharedBlk {
    v4f tile[2][TILE_ROWS * ROW4];   // 2 x 64 KB ping-pong, struct offset 0
    int pref[MAXSEQ + 1];            // chunk-count prefix sum
    int tok0[MAXSEQ];                // cu_seqlens[s]
};

__device__ __forceinline__ int chunks_of(int len, int ks, int st) {
    return (len >= ks) ? (len - ks) / st + 1 : 0;
}

// One TDM descriptor: DMA `rows` contiguous token rows (rows x 2048B 2D tile,
// dim0 = 512 f32 elements, stride 512) from global into LDS at `lds_off`.
__device__ __forceinline__ void tdm_load(unsigned lds_off, const float* src, int rows) {
    unsigned long long ga = (unsigned long long)(const void*)src;

    v4u g0;
    g0.x = 1u;                                  // count=1, user mode, no gather
    g0.y = lds_off;                             // lds_addr (bytes)
    g0.z = (unsigned)ga;                        // global_addr[31:0]
    g0.w = (unsigned)((ga >> 32) & 0x1FFFFFFu)  // global_addr[56:32]
         | (2u << 30);                          // type = 2 ("image")

    v8i g1;
    g1[0] = (int)(2u << 16);                    // wg_mask=0, data_size=2 (4B)
    g1[1] = (int)(((unsigned)ROW & 0xFFFFu) << 16);          // tensor_dim0[15:0]
    g1[2] = (int)((((unsigned)ROW >> 16) & 0xFFFFu)          // tensor_dim0[31:16]
                | (((unsigned)rows & 0xFFFFu) << 16));       // tensor_dim1[15:0]
    g1[3] = (int)((((unsigned)rows >> 16) & 0xFFFFu)         // tensor_dim1[31:16]
                | (((unsigned)ROW & 0xFFFFu) << 16));        // tile_dim0 = 512
    g1[4] = (int)((unsigned)rows & 0xFFFFu);    // tile_dim1 = rows, tile_dim2 = 0
    g1[5] = (int)ROW;                           // tensor_dim0_stride[31:0] = 512
    g1[6] = 0;
    g1[7] = 0;

    v4i g2 = (v4i){0, 0, 0, 0};                 // <=2D tensor: groups 2/3 unused
    v4i g3 = (v4i){0, 0, 0, 0};
    v8i g4 = (v8i){0, 0, 0, 0, 0, 0, 0, 0};

    __builtin_amdgcn_tensor_load_to_lds(g0, g1, g2, g3, g4, 0);   // TENSORcnt++
}

__global__ __launch_bounds__(THREADS)
void CompressK_59691455480627_kernel(const float* __restrict__ k,
                                     const int*   __restrict__ cu,
                                     const int*   __restrict__ p_ks,
                                     const int*   __restrict__ p_st,
                                     float*       __restrict__ out,
                                     int nseq) {
    __shared__ SharedBlk sh;

    const int ks  = *p_ks;
    const int st  = *p_st;
    const int tid = threadIdx.x;
    if (nseq > MAXSEQ) nseq = MAXSEQ;   // LDS metadata capacity (ref batch: 4)

    // ---- per-sequence chunk prefix: one division per sequence, once per block
    for (int s = tid; s < nseq; s += THREADS) {
        sh.tok0[s]     = cu[s];
        sh.pref[s + 1] = chunks_of(cu[s + 1] - cu[s], ks, st);
    }
    if (tid == 0) sh.pref[0] = 0;
    __syncthreads();
    if (tid == 0)
        for (int s = 0; s < nseq; ++s) sh.pref[s + 1] += sh.pref[s];
    __syncthreads();
    const int total = sh.pref[nseq];

    // tail of d_out: cu_seqlens_compressed (int32) after total*ROW floats
    if (blockIdx.x == 0 && tid == 0) {
        int* tail = (int*)(out + (size_t)total * ROW);
        for (int s = 0; s <= nseq; ++s) tail[s] = sh.pref[s];
    }

    const float inv = 1.0f / (float)ks;
    const int   stride = (int)gridDim.x;

    if (ks <= TILE_ROWS) {
        // ======== pipelined path: one tile per chunk, ping-pong DMA ========
        int g = (int)blockIdx.x;
        if (g < total && tid == 0) {
            int s = 0; while (sh.pref[s + 1] <= g) ++s;
            int tok0 = sh.tok0[s] + (g - sh.pref[s]) * st;
            tdm_load(0u, k + (size_t)tok0 * ROW, ks);           // prologue DMA -> buf 0
        }
        int p = 0;
        for (; g < total; g += stride, p ^= 1) {
            const int gn = g + stride;
            if (tid == 0) {
                if (gn < total) {
                    int s = 0; while (sh.pref[s + 1] <= gn) ++s;
                    int tok0 = sh.tok0[s] + (gn - sh.pref[s]) * st;
                    tdm_load(p ? 0u : (unsigned)TILE_BYTES,      // next chunk -> other buf
                             k + (size_t)tok0 * ROW, ks);
                    __builtin_amdgcn_s_wait_tensorcnt(1);        // older DMA (buf p) done
                } else {
                    __builtin_amdgcn_s_wait_tensorcnt(0);
                }
            }
            __syncthreads();                 // publish buf p to all 4 waves

            v4f acc = (v4f){0.f, 0.f, 0.f, 0.f};
            if (ks == TILE_ROWS) {
                #pragma unroll
                for (int r = 0; r < TILE_ROWS; ++r)
                    acc += sh.tile[p][r * ROW4 + tid];           // ds_load_b128
            } else {
                for (int r = 0; r < ks; ++r)
                    acc += sh.tile[p][r * ROW4 + tid];
            }

            __builtin_nontemporal_store(acc * inv, (v4f*)(out + (size_t)g * ROW) + tid);
            __syncthreads();                 // buf p readers done before it is re-targeted
        }
    } else {
        // ======== serial path: chunks larger than one 64 KB tile ========
        for (int g = (int)blockIdx.x; g < total; g += stride) {
            int s = 0; while (sh.pref[s + 1] <= g) ++s;
            const int tok0 = sh.tok0[s] + (g - sh.pref[s]) * st;

            v4f acc = (v4f){0.f, 0.f, 0.f, 0.f};
            for (int r0 = 0; r0 < ks; r0 += TILE_ROWS) {
                const int rows = (ks - r0 < TILE_ROWS) ? (ks - r0) : TILE_ROWS;
                __syncthreads();
                if (tid == 0) {
                    tdm_load(0u, k + (size_t)(tok0 + r0) * ROW, rows);
                    __builtin_amdgcn_s_wait_tensorcnt(0);
                }
                __syncthreads();
                for (int r = 0; r < rows; ++r)
                    acc += sh.tile[0][r * ROW4 + tid];
            }
            __builtin_nontemporal_store(acc * inv, (v4f*)(out + (size_t)g * ROW) + tid);
            __syncthreads();
        }
    }
}

extern "C" void kernel_launch(void* const* d_in, const int* in_sizes, int n_in,
                              void* d_out, int out_size, void* d_ws, size_t ws_size,
                              hipStream_t stream) {
    (void)n_in; (void)out_size; (void)d_ws; (void)ws_size;
    const float* k   = (const float*)d_in[0];
    const int*   cu  = (const int*)d_in[1];
    const int*   ks  = (const int*)d_in[2];
    const int*   st  = (const int*)d_in[3];
    float*       out = (float*)d_out;
    const int nseq = in_sizes[1] - 1;

    // Persistent grid: chunk count lives on-device (ks/stride are device scalars),
    // so grid-stride over a fixed grid.
    CompressK_59691455480627_kernel<<<dim3(1024), dim3(THREADS), 0, stream>>>(
        k, cu, ks, st, out, nseq);
}